// De_KL_39865886442001
// MI455X (gfx1250) — compile-verified
//
#include <hip/hip_runtime.h>

typedef __attribute__((ext_vector_type(2))) float v2f;
typedef __attribute__((ext_vector_type(8))) float v8f;

#define N_PTS 2048
#define DDIM  16
#define KPAD  36                      // 33 features (z^2[16], z[16], 1) padded to 36
#define LN2PI 1.8378770457940425f     // log(2*3.1415926), matches reference constant

// ---------------------------------------------------------------------------
// Phase A: per-(j,d) coefficients + WMMA panels + log_pz row sums.
//   p[i,j,d] = nhw[j,d]*(z[i,d]-mu[j,d])^2 + b2[j,d]
//   S[i,j]   = sum_k A[i,k]*B[k,j],  A=[z^2, z, 1], B=[nhw, w*mu, C_j]
// One wave handles 2 rows (lane%16 = d, lane/16 = row parity).
// ---------------------------------------------------------------------------
__global__ void __launch_bounds__(256) prep_kernel(
    const float* __restrict__ z, const float* __restrict__ mu,
    const float* __restrict__ logvar,
    float* __restrict__ Apanel, float* __restrict__ Bpanel,
    float* __restrict__ nhwArr, float* __restrict__ b2Arr,
    float* __restrict__ lpz) {
  const int lane = threadIdx.x & 31;
  const int wv   = threadIdx.x >> 5;
  const int d    = lane & 15;
  const int hi   = lane >> 4;
  const int j    = blockIdx.x * 16 + wv * 2 + hi;

  const float lv = logvar[j * DDIM + d];
  const float m  = mu[j * DDIM + d];
  const float zv = z[j * DDIM + d];

  const float w   = __expf(-lv);
  const float nhw = -0.5f * w;
  const float wm  = w * m;
  const float b2  = -0.5f * (lv + LN2PI);

  nhwArr[j * DDIM + d] = nhw;
  b2Arr [j * DDIM + d] = b2;

  Bpanel[j * KPAD + d]      = nhw;     // B rows k=0..15
  Bpanel[j * KPAD + 16 + d] = wm;      // B rows k=16..31
  Apanel[j * KPAD + d]      = zv * zv; // A cols k=0..15
  Apanel[j * KPAD + 16 + d] = zv;      // A cols k=16..31

  float cel = __builtin_fmaf(nhw, m * m, b2);  // per-(j,d) constant term
  float lzt = zv * zv;
  #pragma unroll
  for (int off = 1; off < 16; off <<= 1) {     // reduce over d within half-wave
    cel += __shfl_xor(cel, off, 32);
    lzt += __shfl_xor(lzt, off, 32);
  }
  if (d < 4) {                                  // k=32 value, k=33..35 zero pad
    Bpanel[j * KPAD + 32 + d] = (d == 0) ? cel  : 0.0f;
    Apanel[j * KPAD + 32 + d] = (d == 0) ? 1.0f : 0.0f;
  }
  if (d == 0) lpz[j] = -0.5f * (lzt + (float)DDIM * LN2PI);
}

// ---------------------------------------------------------------------------
// Phase B: one block owns a 16-row i-tile. 8 waves stride over 128 j-tiles.
// Per j-tile:  S via 9x V_WMMA_F32_16X16X4_F32, then elementwise exp(-|p|).
// Deterministic cross-wave combine through per-wave LDS slots (no atomics).
// ---------------------------------------------------------------------------
__global__ void __launch_bounds__(256) main_kernel(
    const float* __restrict__ z, const float* __restrict__ mu,
    const float* __restrict__ Apanel, const float* __restrict__ Bpanel,
    const float* __restrict__ nhwArr, const float* __restrict__ b2Arr,
    float* __restrict__ partials) {
  __shared__ float t2s[8][256];   // per-wave t2 partials  [wave][(i_local)*16 + d]
  __shared__ float t1s[8][16];    // per-wave t1 partials  [wave][i_local]
  __shared__ float sMs[8];        // per-wave sum(relu(S))
  __shared__ float red[256];      // log terms for t2
  __shared__ float lqs[16];       // log terms for t1

  const int lane = threadIdx.x & 31;
  const int wv   = threadIdx.x >> 5;
  const int n    = lane & 15;     // WMMA col / feature dim d
  const int hi   = lane >> 4;     // lane half
  const int i0   = blockIdx.x * 16;

  // A operand for this i-tile: lane m=n holds A[m][k0+2*hi], A[m][k0+2*hi+1]
  v2f areg[9];
  #pragma unroll
  for (int s = 0; s < 9; ++s)
    areg[s] = *(const v2f*)(Apanel + (size_t)(i0 + n) * KPAD + 4 * s + 2 * hi);

  // z values for elementwise pass: lane holds (d=n, row=i0+2*ii+hi)
  float zreg[8];
  #pragma unroll
  for (int ii = 0; ii < 8; ++ii)
    zreg[ii] = z[(i0 + 2 * ii + hi) * DDIM + n];

  float t1a[8] = {};
  float t2a[8] = {};
  float sumM = 0.0f;

  for (int jt = wv; jt < N_PTS / 16; jt += 8) {
    const int j0 = jt * 16;

    // B operand: lane col n holds B[k0+2*hi][j0+n] pair (Bpanel is [j][k])
    v2f breg[9];
    #pragma unroll
    for (int s = 0; s < 9; ++s)
      breg[s] = *(const v2f*)(Bpanel + (size_t)(j0 + n) * KPAD + 4 * s + 2 * hi);

    v8f c = {};
    #pragma unroll
    for (int s = 0; s < 9; ++s)
      c = __builtin_amdgcn_wmma_f32_16x16x4_f32(
          false, areg[s], false, breg[s], (short)0, c, false, false);

    // c[v] = S[i0 + v + 8*hi][j0 + n]
    #pragma unroll
    for (int v = 0; v < 8; ++v) {
      const float sv = c[v];
      sumM  += fmaxf(sv, 0.0f);
      t1a[v] += __expf(-__builtin_fabsf(sv));   // exp(-M)+exp(S-M) = 1+exp(-|S|)
    }

    // elementwise t2: lane=(d, parity), 8 i-rows in registers
    for (int jj = 0; jj < 16; ++jj) {
      const int j = j0 + jj;
      const float muv = mu[j * DDIM + n];
      const float nh  = nhwArr[j * DDIM + n];
      const float bb  = b2Arr[j * DDIM + n];
      #pragma unroll
      for (int ii = 0; ii < 8; ++ii) {
        const float diff = zreg[ii] - muv;
        const float p = __builtin_fmaf(nh, diff * diff, bb);
        t2a[ii] += __expf(-__builtin_fabsf(p)); // 1+exp(-|p|), "+1" folded as N later
      }
    }
  }

  // intra-wave reductions, stash per-wave results
  #pragma unroll
  for (int v = 0; v < 8; ++v) {
    float t = t1a[v];
    t += __shfl_xor(t, 1, 32); t += __shfl_xor(t, 2, 32);
    t += __shfl_xor(t, 4, 32); t += __shfl_xor(t, 8, 32);
    if (n == 0) t1s[wv][v + 8 * hi] = t;
  }
  {
    float t = sumM;
    t += __shfl_xor(t, 1, 32);  t += __shfl_xor(t, 2, 32);
    t += __shfl_xor(t, 4, 32);  t += __shfl_xor(t, 8, 32);
    t += __shfl_xor(t, 16, 32);
    if (lane == 0) sMs[wv] = t;
  }
  #pragma unroll
  for (int ii = 0; ii < 8; ++ii)
    t2s[wv][(2 * ii + hi) * 16 + n] = t2a[ii];  // unique (i,d) per lane

  __syncthreads();

  // deterministic cross-wave combine (fixed order)
  {
    const int tid = threadIdx.x;   // r = tid/16, d = tid%16
    float t2sum = 0.0f;
    #pragma unroll
    for (int w2 = 0; w2 < 8; ++w2) t2sum += t2s[w2][tid];
    red[tid] = __logf((float)N_PTS + t2sum + 1e-5f);
    if (tid < 16) {
      float t1sum = 0.0f;
      #pragma unroll
      for (int w2 = 0; w2 < 8; ++w2) t1sum += t1s[w2][tid];
      lqs[tid] = __logf((float)N_PTS + t1sum + 1e-5f);
    }
  }
  __syncthreads();

  if (threadIdx.x == 0) {
    float slqp = 0.0f, slq = 0.0f, sM = 0.0f;
    for (int t = 0; t < 256; ++t) slqp += red[t];   // sum_d log(t2) over 16 rows
    for (int r = 0; r < 16; ++r)  slq  += lqs[r];
    for (int w2 = 0; w2 < 8; ++w2) sM  += sMs[w2];
    partials[blockIdx.x * 3 + 0] = sM;
    partials[blockIdx.x * 3 + 1] = slq;
    partials[blockIdx.x * 3 + 2] = slqp;
  }
}

// ---------------------------------------------------------------------------
// Final: fixed-order reduction of 128 block partials + 2048 lpz rows -> 3 outs
// ---------------------------------------------------------------------------
__global__ void final_kernel(const float* __restrict__ partials,
                             const float* __restrict__ lpz,
                             const int* __restrict__ beta,
                             float* __restrict__ out) {
  const int lane = threadIdx.x;
  float sM = 0.0f, slq = 0.0f, slqp = 0.0f, slpz = 0.0f;
  for (int b = lane; b < N_PTS / 16; b += 32) {
    sM   += partials[b * 3 + 0];
    slq  += partials[b * 3 + 1];
    slqp += partials[b * 3 + 2];
  }
  for (int i = lane; i < N_PTS; i += 32) slpz += lpz[i];
  #pragma unroll
  for (int off = 1; off < 32; off <<= 1) {
    sM   += __shfl_xor(sM, off, 32);
    slq  += __shfl_xor(slq, off, 32);
    slqp += __shfl_xor(slqp, off, 32);
    slpz += __shfl_xor(slpz, off, 32);
  }
  if (lane == 0) {
    const float nf = (float)N_PTS;
    const float n3 = nf * nf * nf;
    const float bf = (float)beta[0];
    out[0] = sM / n3 - slq / nf;          // idx_code_mi
    out[1] = bf * (slq - slqp) / nf;      // total_corr * beta
    out[2] = (slqp - slpz) / nf;          // dim_wise_kl
  }
}

extern "C" void kernel_launch(void* const* d_in, const int* in_sizes, int n_in,
                              void* d_out, int out_size, void* d_ws, size_t ws_size,
                              hipStream_t stream) {
  const float* z      = (const float*)d_in[0];
  const float* mu     = (const float*)d_in[1];
  const float* logvar = (const float*)d_in[2];
  const int*   beta   = (const int*)d_in[3];
  float* out = (float*)d_out;

  float* Apanel   = (float*)d_ws;             // N*36
  float* Bpanel   = Apanel + N_PTS * KPAD;    // N*36
  float* nhwArr   = Bpanel + N_PTS * KPAD;    // N*16
  float* b2Arr    = nhwArr + N_PTS * DDIM;    // N*16
  float* lpz      = b2Arr  + N_PTS * DDIM;    // N
  float* partials = lpz + N_PTS;              // 128*3

  prep_kernel<<<N_PTS / 16, 256, 0, stream>>>(z, mu, logvar, Apanel, Bpanel,
                                              nhwArr, b2Arr, lpz);
  main_kernel<<<N_PTS / 16, 256, 0, stream>>>(z, mu, Apanel, Bpanel,
                                              nhwArr, b2Arr, partials);
  final_kernel<<<1, 32, 0, stream>>>(partials, lpz, beta, out);
}